// MultiHeadAttentionLayer_44487271252167
// MI455X (gfx1250) — compile-verified
//
#include <hip/hip_runtime.h>
#include <math.h>

// ---------------------------------------------------------------------------
// MultiHeadAttention graph layer for MI455X (gfx1250, wave32)
//   N=100000 nodes, E=1600000 edges, IN_DIM=128, HEADS=8, OUT_DIM=8 (d=64)
//
//  K1: Q/K/V projections via V_WMMA_F32_16X16X4_F32 (exact fp32, tensor path)
//  K2: per-edge attention score + atomic scatter (memory/atomic bound, ~75us
//      of HBM traffic at 23.3 TB/s; accumulators fit in 192MB L2)
//  K3: normalize wV / (Z + 1e-6)
// ---------------------------------------------------------------------------

typedef __attribute__((ext_vector_type(2))) float v2f;
typedef __attribute__((ext_vector_type(8))) float v8f;

#define IN_DIM 128
#define D_TOT  64    // HEADS * OUT_DIM
#define HEADS  8

// ---- Kernel 1: fused QKV projection, one 16x16 tile per wave --------------
// grid.x = N/16 row tiles, grid.y = 3 (Q,K,V), block = 128 (4 waves = 4 col tiles)
__global__ __launch_bounds__(128) void qkv_proj_wmma(
    const float* __restrict__ x,
    const float* __restrict__ Wq, const float* __restrict__ bq,
    const float* __restrict__ Wk, const float* __restrict__ bk,
    const float* __restrict__ Wv, const float* __restrict__ bv,
    float* __restrict__ Qh, float* __restrict__ Kh, float* __restrict__ Vh,
    int nrows)
{
    const int which = blockIdx.y;
    const float* __restrict__ W   = (which == 0) ? Wq : (which == 1) ? Wk : Wv;
    const float* __restrict__ bia = (which == 0) ? bq : (which == 1) ? bk : bv;
    float* __restrict__ out       = (which == 0) ? Qh : (which == 1) ? Kh : Vh;

    const int wave = threadIdx.x >> 5;   // 0..3 : 16-wide column tile
    const int lane = threadIdx.x & 31;
    const int half = lane >> 4;          // 0: lanes 0-15, 1: lanes 16-31
    const int l16  = lane & 15;

    const int rowBase = blockIdx.x * 16; // N=100000 is an exact multiple of 16
    const int col     = wave * 16 + l16;

    const float* __restrict__ xrow = x + (size_t)(rowBase + l16) * IN_DIM;

    v8f c = {};
    // ISA 16x4 fp32 A layout: lanes 0-15 hold K={0,1}, lanes 16-31 hold K={2,3}
    // B 4x16 layout: VGPR j, lanes 0-15 -> K=j, lanes 16-31 -> K=j+2, N=lane%16
    #pragma unroll 8
    for (int s = 0; s < IN_DIM / 4; ++s) {
        const int kb = 4 * s + 2 * half;
        v2f a;
        a.x = xrow[kb + 0];
        a.y = xrow[kb + 1];
        v2f b;
        b.x = W[(size_t)(kb + 0) * D_TOT + col];
        b.y = W[(size_t)(kb + 1) * D_TOT + col];
        // 8 args: (neg_a, A, neg_b, B, c_mod, C, reuse_a, reuse_b)
        c = __builtin_amdgcn_wmma_f32_16x16x4_f32(
                false, a, false, b, (short)0, c, false, false);
    }

    const float bv0 = bia[col];
    #pragma unroll
    for (int v = 0; v < 8; ++v) {
        const int m = v + 8 * half;      // C/D layout: VGPR v -> row v (+8 for hi lanes)
        out[(size_t)(rowBase + m) * D_TOT + col] = c[v] + bv0;
    }
}

// ---- Kernel 2: per-edge score + scatter -----------------------------------
// 64 threads per edge: thread t = h*8 + d. 8-lane butterfly reduce (wave32,
// groups of 8 stay inside the wave) for the per-head dot product.
__global__ __launch_bounds__(256) void edge_scatter(
    const float* __restrict__ Qh, const float* __restrict__ Kh,
    const float* __restrict__ Vh,
    const float* __restrict__ edge_attr,
    const long long* __restrict__ edge_index,   // int64 [2, E]
    const float* __restrict__ We, const float* __restrict__ be,
    float* __restrict__ wV, float* __restrict__ Z, int E)
{
    const long long g = (long long)blockIdx.x * 256 + threadIdx.x;
    const long long e = g >> 6;
    if (e >= (long long)E) return;
    const int t = (int)(g & 63);
    const int h = t >> 3;
    const int d = t & 7;

    const long long src = edge_index[e];
    const long long dst = edge_index[(long long)E + e];

    const float ev = edge_attr[e] * We[t] + be[t];         // Eh[e, h, d]
    float p = Kh[src * D_TOT + t] * Qh[dst * D_TOT + t]
              * 0.35355339059327373f * ev;                 // 1/sqrt(8)
    // sum over d (8 lanes)
    p += __shfl_xor(p, 1, 8);
    p += __shfl_xor(p, 2, 8);
    p += __shfl_xor(p, 4, 8);
    const float sc = __expf(fminf(fmaxf(p, -5.0f), 5.0f));

    atomicAdd(&wV[dst * D_TOT + t], Vh[src * D_TOT + t] * sc);
    if (d == 0) atomicAdd(&Z[dst * HEADS + h], sc);
}

// ---- Kernel 3: out = wV / (Z + 1e-6) --------------------------------------
__global__ __launch_bounds__(256) void finalize_div(
    const float* __restrict__ wV, const float* __restrict__ Z,
    float* __restrict__ out, int n64)
{
    const int g = blockIdx.x * 256 + threadIdx.x;
    if (g >= n64) return;
    const int node = g >> 6;
    const int h    = (g >> 3) & 7;
    out[g] = wV[g] / (Z[node * HEADS + h] + 1e-6f);
}

// ---------------------------------------------------------------------------
extern "C" void kernel_launch(void* const* d_in, const int* in_sizes, int n_in,
                              void* d_out, int out_size, void* d_ws, size_t ws_size,
                              hipStream_t stream) {
    const float*     x  = (const float*)d_in[0];
    const float*     ea = (const float*)d_in[1];
    const long long* ei = (const long long*)d_in[2];
    const float*     Wq = (const float*)d_in[3];
    const float*     bq = (const float*)d_in[4];
    const float*     Wk = (const float*)d_in[5];
    const float*     bk = (const float*)d_in[6];
    const float*     We = (const float*)d_in[7];
    const float*     be = (const float*)d_in[8];
    const float*     Wv = (const float*)d_in[9];
    const float*     bv = (const float*)d_in[10];
    float*           out = (float*)d_out;

    const int n = in_sizes[0] / IN_DIM;   // 100000
    const int E = in_sizes[1];            // 1600000 (edge_attr is [E,1])
    const size_t n64 = (size_t)n * D_TOT;

    float* ws = (float*)d_ws;
    float* Qh = ws;
    float* Kh = Qh + n64;
    float* Vh = Kh + n64;
    float* wV = Vh + n64;
    float* Z  = wV + n64;                 // n * HEADS floats

    // zero the accumulators every call (deterministic; capture-legal)
    hipMemsetAsync(wV, 0, (n64 + (size_t)n * HEADS) * sizeof(float), stream);

    dim3 gq((n + 15) / 16, 3);
    qkv_proj_wmma<<<gq, 128, 0, stream>>>(x, Wq, bq, Wk, bk, Wv, bv,
                                          Qh, Kh, Vh, n);

    const long long tot = (long long)E * 64;
    const int eblocks = (int)((tot + 255) / 256);
    edge_scatter<<<eblocks, 256, 0, stream>>>(Qh, Kh, Vh, ea, ei, We, be,
                                              wV, Z, E);

    finalize_div<<<((int)n64 + 255) / 256, 256, 0, stream>>>(wV, Z, out, (int)n64);
}